// NetConv_Swi_cts_57939108823644
// MI455X (gfx1250) — compile-verified
//
#include <hip/hip_runtime.h>
#include <hip/hip_bf16.h>

// ---------------------------------------------------------------------------
// NetConv GNN pass for MI455X (gfx1250, wave32).
// - All small GEMMs on V_WMMA_F32_16X16X4_F32 (f32-exact), 16-row tiles/wave.
// - Edge-feature gathers use GLOBAL_LOAD_ASYNC_TO_LDS_B128 (ASYNCcnt path):
//   per-lane gather straight into LDS, fenced with s_wait_asynccnt 0.
// - Segment sums via global f32 atomics (new_nf etc. are L2-resident: 51MB
//   against 192MB L2).
// ---------------------------------------------------------------------------

typedef __attribute__((ext_vector_type(2))) float v2f;
typedef __attribute__((ext_vector_type(8))) float v8f;

#define DEVINL __device__ __forceinline__

DEVINL void wavebar() { __builtin_amdgcn_wave_barrier(); }

DEVINL float sigmoidf(float x) { return 1.0f / (1.0f + __expf(-x)); }

// Generic LDS pointer -> 32-bit LDS byte offset (generic LDS addr is
// {SHARED_BASE[63:32], offset[31:0]} per the CDNA5 aperture rules).
DEVINL unsigned lds_off(const void* p) { return (unsigned)(size_t)p; }

// Async per-lane 16B gather: copies 16 bytes from global base plus gOffBytes to the LDS address ldsAddr. ASYNCcnt.
DEVINL void async_gather16(unsigned ldsAddr, unsigned gOffBytes, const void* base) {
  asm volatile("global_load_async_to_lds_b128 %0, %1, %2"
               :: "v"(ldsAddr), "v"(gOffBytes), "s"(base)
               : "memory");
}
DEVINL void wait_async0() { asm volatile("s_wait_asynccnt 0" ::: "memory"); }

// ---- block-cooperative weight staging into LDS ----------------------------
DEVINL void stage(float* dst, const float* src, int n) {
  for (int i = threadIdx.x; i < n; i += blockDim.x) dst[i] = src[i];
}
DEVINL void stage_pad_cols(float* dst, const float* src, int K, int ncol, int npad) {
  for (int i = threadIdx.x; i < K * npad; i += blockDim.x) {
    int k = i / npad, n = i - k * npad;
    dst[i] = (n < ncol) ? src[k * ncol + n] : 0.0f;
  }
}
DEVINL void stage_pad_vec(float* dst, const float* src, int ncol, int npad) {
  for (int i = threadIdx.x; i < npad; i += blockDim.x)
    dst[i] = (i < ncol) ? src[i] : 0.0f;
}

// ---------------------------------------------------------------------------
// Wave-level 16xK @ KxN16 f32 GEMM using V_WMMA_F32_16X16X4_F32.
//   act : LDS activation tile, row-major [16][sIn] (rows = edges/nodes)
//   W   : LDS weights, row-major [K][wN]; uses columns colBase..colBase+15
//   out : LDS output tile, row-major [16][sOut], cols colBase..colBase+15
// A-fragment (16x4 f32): lane l (m=l&15, h=l>>4): {A[m][4s+2h], A[m][4s+2h+1]}
// B-fragment (4x16 f32): lane l (n=l&15):         {B[4s+2h][n],  B[4s+2h+1][n]}
// C/D (16x16 f32): VGPR r: lanes 0-15 -> M=r, lanes 16-31 -> M=8+r; N=lane&15
// ---------------------------------------------------------------------------
template <int K>
DEVINL void wave_gemm16(const float* act, int sIn,
                        const float* W, int wN, int colBase, const float* bias,
                        float* out, int sOut, bool lrelu) {
  const int l = threadIdx.x & 31;
  const int m = l & 15;
  const int h = l >> 4;
  v8f acc = {0.f, 0.f, 0.f, 0.f, 0.f, 0.f, 0.f, 0.f};
  wavebar();
#pragma unroll
  for (int s = 0; s < K / 4; ++s) {
    const int k = s * 4 + h * 2;
    v2f a = {act[m * sIn + k], act[m * sIn + k + 1]};
    v2f b = {W[k * wN + colBase + m], W[(k + 1) * wN + colBase + m]};
    acc = __builtin_amdgcn_wmma_f32_16x16x4_f32(false, a, false, b,
                                                (short)0, acc, false, false);
  }
  const float bv = bias[colBase + m];
#pragma unroll
  for (int r = 0; r < 8; ++r) {
    float y = acc[r] + bv;
    if (lrelu) y = fmaxf(y, 0.2f * y);  // leaky_relu(0.2)
    out[(r + 8 * h) * sOut + colBase + m] = y;
  }
  wavebar();
}

// ---------------------------------------------------------------------------
__global__ void k_zero(float* __restrict__ p, size_t n) {
  size_t i = (size_t)blockIdx.x * blockDim.x + threadIdx.x;
  size_t st = (size_t)gridDim.x * blockDim.x;
  for (; i < n; i += st) p[i] = 0.0f;
}

__global__ void k_mask(const int* __restrict__ pins, int* __restrict__ mask, int P) {
  int i = blockIdx.x * blockDim.x + threadIdx.x;
  if (i < P) mask[pins[i]] = 1;
}

// ---------------------------------------------------------------------------
// Pass 1: efi = MLP_msg_o2i([nf[src], nf[dst], nef_out]) ; new_nf += scatter
// sizes: 144 -> 16 -> 16 -> 16 -> 64, leaky relu between.
// ---------------------------------------------------------------------------
__global__ void __launch_bounds__(128) k_msg_o2i(
    const float* __restrict__ nf, const float* __restrict__ nef_out,
    const int* __restrict__ src, const int* __restrict__ dst,
    const float* W0, const float* B0, const float* W1, const float* B1,
    const float* W2, const float* B2, const float* W3, const float* B3,
    float* __restrict__ new_nf, int E) {
  __shared__ float sW0[144 * 16], sB0[16], sW1[256], sB1[16];
  __shared__ float sW2[256], sB2[16], sW3[16 * 64], sB3[64];
  __shared__ float sA[4][16 * 148];
  __shared__ float sBf[4][16 * 20];
  stage(sW0, W0, 144 * 16); stage(sB0, B0, 16);
  stage(sW1, W1, 256);      stage(sB1, B1, 16);
  stage(sW2, W2, 256);      stage(sB2, B2, 16);
  stage(sW3, W3, 16 * 64);  stage(sB3, B3, 64);
  __syncthreads();
  const int wave = threadIdx.x >> 5;
  const int l = threadIdx.x & 31, m = l & 15, h = l >> 4;
  float* bufA = sA[wave];
  float* bufB = sBf[wave];
  const unsigned aBase = lds_off(bufA) + (unsigned)(m * 148 * 4);
  const int nTiles = (E + 15) >> 4;
  for (int tile = blockIdx.x * 4 + wave; tile < nTiles; tile += gridDim.x * 4) {
    const int base = tile << 4;
    int e = base + m; if (e >= E) e = E - 1;
    const unsigned so = (unsigned)src[e] * 256u;   // byte offset into nf
    const unsigned dofs = (unsigned)dst[e] * 256u;
    // async gather 16 edges x 144 feats into LDS; source region uniform per c
#pragma unroll
    for (int c = 0; c < 8; ++c) {               // f in [0,64): nf[src]
      const unsigned f = (unsigned)((2 * c + h) << 2);
      async_gather16(aBase + f * 4u, so + f * 4u, nf);
    }
#pragma unroll
    for (int c = 8; c < 16; ++c) {              // f in [64,128): nf[dst]
      const unsigned f = (unsigned)((2 * c + h) << 2);
      async_gather16(aBase + f * 4u, dofs + (f - 64u) * 4u, nf);
    }
#pragma unroll
    for (int c = 16; c < 18; ++c) {             // f in [128,144): nef_out[e]
      const unsigned f = (unsigned)((2 * c + h) << 2);
      async_gather16(aBase + f * 4u, (unsigned)e * 64u + (f - 128u) * 4u, nef_out);
    }
    wait_async0();
    wave_gemm16<144>(bufA, 148, sW0, 16, 0, sB0, bufB, 20, true);
    wave_gemm16<16>(bufB, 20, sW1, 16, 0, sB1, bufA, 20, true);
    wave_gemm16<16>(bufA, 20, sW2, 16, 0, sB2, bufB, 20, true);
#pragma unroll
    for (int t = 0; t < 4; ++t)
      wave_gemm16<16>(bufB, 20, sW3, 64, t * 16, sB3, bufA, 68, false);
    // scatter-add 16x64 tile into new_nf[dst]
    for (int i = l; i < 16 * 64; i += 32) {
      const int r = i >> 6, c = i & 63;
      const int ee = base + r;
      if (ee < E) atomicAdd(new_nf + (size_t)dst[ee] * 64 + c, bufA[r * 68 + c]);
    }
    wavebar();
  }
}

// ---------------------------------------------------------------------------
// node_matrix colsums: MLP_net_in_trans1(new_nf): 64 -> 16 -> 16, then
// cs[j] = sum_i y[4i+j]  (j=0..3)  stored to node_cs[N][4].
// ---------------------------------------------------------------------------
__global__ void __launch_bounds__(128) k_node_trans(
    const float* __restrict__ new_nf,
    const float* W0, const float* B0, const float* W1, const float* B1,
    float* __restrict__ node_cs, int N) {
  __shared__ float sW0[64 * 16], sB0[16], sW1[256], sB1[16];
  __shared__ float sA[4][16 * 68];
  __shared__ float sBf[4][16 * 20];
  stage(sW0, W0, 64 * 16); stage(sB0, B0, 16);
  stage(sW1, W1, 256);     stage(sB1, B1, 16);
  __syncthreads();
  const int wave = threadIdx.x >> 5;
  const int l = threadIdx.x & 31, m = l & 15, h = l >> 4;
  float* bufA = sA[wave];
  float* bufB = sBf[wave];
  const int nTiles = (N + 15) >> 4;
  for (int tile = blockIdx.x * 4 + wave; tile < nTiles; tile += gridDim.x * 4) {
    const int base = tile << 4;
    int n = base + m; if (n >= N) n = N - 1;
#pragma unroll
    for (int c = 0; c < 8; ++c) {
      const int f = (2 * c + h) << 2;
      *(float4*)(bufA + m * 68 + f) = *(const float4*)(new_nf + (size_t)n * 64 + f);
    }
    wave_gemm16<64>(bufA, 68, sW0, 16, 0, sB0, bufB, 20, true);
    wave_gemm16<16>(bufB, 20, sW1, 16, 0, sB1, bufA, 20, false);
    if (h == 0) {
      const int n2 = base + m;
      if (n2 < N) {
#pragma unroll
        for (int j = 0; j < 4; ++j)
          node_cs[(size_t)n2 * 4 + j] = bufA[m * 20 + j] + bufA[m * 20 + 4 + j] +
                                        bufA[m * 20 + 8 + j] + bufA[m * 20 + 12 + j];
      }
    }
    wavebar();
  }
}

// ---------------------------------------------------------------------------
// Attention: em = MLP_edge_mat1([new_nf[src_in], nef_in]) (80->16->16),
// raw_att = sigmoid(dot4(node_cs[dst_in], colsum4(em))); att_sum scatter.
// ---------------------------------------------------------------------------
__global__ void __launch_bounds__(128) k_edge_att(
    const float* __restrict__ new_nf, const float* __restrict__ nef_in,
    const int* __restrict__ src_in, const int* __restrict__ dst_in,
    const float* __restrict__ node_cs,
    const float* W0, const float* B0, const float* W1, const float* B1,
    float* __restrict__ raw_att, float* __restrict__ att_sum, int E) {
  __shared__ float sW0[80 * 16], sB0[16], sW1[256], sB1[16];
  __shared__ float sA[4][16 * 84];
  __shared__ float sBf[4][16 * 20];
  stage(sW0, W0, 80 * 16); stage(sB0, B0, 16);
  stage(sW1, W1, 256);     stage(sB1, B1, 16);
  __syncthreads();
  const int wave = threadIdx.x >> 5;
  const int l = threadIdx.x & 31, m = l & 15, h = l >> 4;
  float* bufA = sA[wave];
  float* bufB = sBf[wave];
  const unsigned aBase = lds_off(bufA) + (unsigned)(m * 84 * 4);
  const int nTiles = (E + 15) >> 4;
  for (int tile = blockIdx.x * 4 + wave; tile < nTiles; tile += gridDim.x * 4) {
    const int base = tile << 4;
    int e = base + m; if (e >= E) e = E - 1;
    const unsigned so = (unsigned)src_in[e] * 256u;
#pragma unroll
    for (int c = 0; c < 8; ++c) {               // f in [0,64): new_nf[src_in]
      const unsigned f = (unsigned)((2 * c + h) << 2);
      async_gather16(aBase + f * 4u, so + f * 4u, new_nf);
    }
#pragma unroll
    for (int c = 8; c < 10; ++c) {              // f in [64,80): nef_in[e]
      const unsigned f = (unsigned)((2 * c + h) << 2);
      async_gather16(aBase + f * 4u, (unsigned)e * 64u + (f - 64u) * 4u, nef_in);
    }
    wait_async0();
    wave_gemm16<80>(bufA, 84, sW0, 16, 0, sB0, bufB, 20, true);
    wave_gemm16<16>(bufB, 20, sW1, 16, 0, sB1, bufA, 20, false);
    if (h == 0) {
      const int e2 = base + m;
      if (e2 < E) {
        const int d = dst_in[e2];
        float dotv = 0.f;
#pragma unroll
        for (int j = 0; j < 4; ++j) {
          float cs = bufA[m * 20 + j] + bufA[m * 20 + 4 + j] +
                     bufA[m * 20 + 8 + j] + bufA[m * 20 + 12 + j];
          dotv += cs * node_cs[(size_t)d * 4 + j];
        }
        float ra = sigmoidf(dotv);
        raw_att[e2] = ra;
        atomicAdd(att_sum + d, ra);
      }
    }
    wavebar();
  }
}

// ---------------------------------------------------------------------------
// x = MLP_msg_i2o([new_nf[src_in], nf[dst_in], nef_in]) (144->16^3->13pad16)
// k=sig(x0); score=raw_att/(att_sum[dst]+1e-3);
// nfo1[dst] += x[1:5]*k*score ; nfo2[dst] += x[5:13]*k
// ---------------------------------------------------------------------------
__global__ void __launch_bounds__(128) k_msg_i2o(
    const float* __restrict__ new_nf, const float* __restrict__ nf,
    const float* __restrict__ nef_in,
    const int* __restrict__ src_in, const int* __restrict__ dst_in,
    const float* __restrict__ raw_att, const float* __restrict__ att_sum,
    const float* W0, const float* B0, const float* W1, const float* B1,
    const float* W2, const float* B2, const float* W3, const float* B3,
    float* __restrict__ nfo1, float* __restrict__ nfo2, int E) {
  __shared__ float sW0[144 * 16], sB0[16], sW1[256], sB1[16];
  __shared__ float sW2[256], sB2[16], sW3[256], sB3[16];
  __shared__ float sA[4][16 * 148];
  __shared__ float sBf[4][16 * 20];
  stage(sW0, W0, 144 * 16); stage(sB0, B0, 16);
  stage(sW1, W1, 256);      stage(sB1, B1, 16);
  stage(sW2, W2, 256);      stage(sB2, B2, 16);
  stage_pad_cols(sW3, W3, 16, 13, 16);  // 16x13 -> 16x16 zero-padded
  stage_pad_vec(sB3, B3, 13, 16);
  __syncthreads();
  const int wave = threadIdx.x >> 5;
  const int l = threadIdx.x & 31, m = l & 15, h = l >> 4;
  float* bufA = sA[wave];
  float* bufB = sBf[wave];
  const unsigned aBase = lds_off(bufA) + (unsigned)(m * 148 * 4);
  const int nTiles = (E + 15) >> 4;
  for (int tile = blockIdx.x * 4 + wave; tile < nTiles; tile += gridDim.x * 4) {
    const int base = tile << 4;
    int e = base + m; if (e >= E) e = E - 1;
    const unsigned so = (unsigned)src_in[e] * 256u;
    const unsigned dofs = (unsigned)dst_in[e] * 256u;
#pragma unroll
    for (int c = 0; c < 8; ++c) {               // f in [0,64): new_nf[src_in]
      const unsigned f = (unsigned)((2 * c + h) << 2);
      async_gather16(aBase + f * 4u, so + f * 4u, new_nf);
    }
#pragma unroll
    for (int c = 8; c < 16; ++c) {              // f in [64,128): nf[dst_in]
      const unsigned f = (unsigned)((2 * c + h) << 2);
      async_gather16(aBase + f * 4u, dofs + (f - 64u) * 4u, nf);
    }
#pragma unroll
    for (int c = 16; c < 18; ++c) {             // f in [128,144): nef_in[e]
      const unsigned f = (unsigned)((2 * c + h) << 2);
      async_gather16(aBase + f * 4u, (unsigned)e * 64u + (f - 128u) * 4u, nef_in);
    }
    wait_async0();
    wave_gemm16<144>(bufA, 148, sW0, 16, 0, sB0, bufB, 20, true);
    wave_gemm16<16>(bufB, 20, sW1, 16, 0, sB1, bufA, 20, true);
    wave_gemm16<16>(bufA, 20, sW2, 16, 0, sB2, bufB, 20, true);
    wave_gemm16<16>(bufB, 20, sW3, 16, 0, sB3, bufA, 20, false);
    {
      const int e2 = base + m;
      if (e2 < E) {
        const int d = dst_in[e2];
        const float kk = sigmoidf(bufA[m * 20 + 0]);
        if (h == 0) {
          const float sc = raw_att[e2] / (att_sum[d] + 0.001f);
          const float ks = kk * sc;
#pragma unroll
          for (int j = 0; j < 4; ++j)
            atomicAdd(nfo1 + (size_t)d * 4 + j, bufA[m * 20 + 1 + j] * ks);
        } else {
#pragma unroll
          for (int j = 0; j < 8; ++j)
            atomicAdd(nfo2 + (size_t)d * 8 + j, bufA[m * 20 + 5 + j] * kk);
        }
      }
    }
    wavebar();
  }
}

// ---------------------------------------------------------------------------
// reduced = MLP_reduce_o([new_nf, nfo1, nfo2]) (76->16^3->64)
// out = mask ? reduced : new_nf
// ---------------------------------------------------------------------------
__global__ void __launch_bounds__(128) k_reduce(
    const float* __restrict__ new_nf, const float* __restrict__ nfo1,
    const float* __restrict__ nfo2, const int* __restrict__ mask,
    const float* W0, const float* B0, const float* W1, const float* B1,
    const float* W2, const float* B2, const float* W3, const float* B3,
    float* __restrict__ out, int N) {
  __shared__ float sW0[76 * 16], sB0[16], sW1[256], sB1[16];
  __shared__ float sW2[256], sB2[16], sW3[16 * 64], sB3[64];
  __shared__ float sA[4][16 * 80];
  __shared__ float sBf[4][16 * 20];
  stage(sW0, W0, 76 * 16); stage(sB0, B0, 16);
  stage(sW1, W1, 256);     stage(sB1, B1, 16);
  stage(sW2, W2, 256);     stage(sB2, B2, 16);
  stage(sW3, W3, 16 * 64); stage(sB3, B3, 64);
  __syncthreads();
  const int wave = threadIdx.x >> 5;
  const int l = threadIdx.x & 31, m = l & 15, h = l >> 4;
  float* bufA = sA[wave];
  float* bufB = sBf[wave];
  const int nTiles = (N + 15) >> 4;
  for (int tile = blockIdx.x * 4 + wave; tile < nTiles; tile += gridDim.x * 4) {
    const int base = tile << 4;
    int n = base + m; if (n >= N) n = N - 1;
    // 76 feats = 64 new_nf + 4 nfo1 + 8 nfo2  (19 float4 chunks)
#pragma unroll
    for (int c = 0; c < 10; ++c) {
      const int cc = 2 * c + h;
      if (cc < 19) {
        const int f = cc << 2;
        float4 v;
        if (f < 64)       v = *(const float4*)(new_nf + (size_t)n * 64 + f);
        else if (f == 64) v = *(const float4*)(nfo1 + (size_t)n * 4);
        else              v = *(const float4*)(nfo2 + (size_t)n * 8 + (f - 68));
        *(float4*)(bufA + m * 80 + f) = v;
      }
    }
    wave_gemm16<76>(bufA, 80, sW0, 16, 0, sB0, bufB, 20, true);
    wave_gemm16<16>(bufB, 20, sW1, 16, 0, sB1, bufA, 20, true);
    wave_gemm16<16>(bufA, 20, sW2, 16, 0, sB2, bufB, 20, true);
#pragma unroll
    for (int t = 0; t < 4; ++t)
      wave_gemm16<16>(bufB, 20, sW3, 64, t * 16, sB3, bufA, 68, false);
    for (int i = l; i < 16 * 64; i += 32) {
      const int r = i >> 6, c = i & 63;
      const int n2 = base + r;
      if (n2 < N)
        out[(size_t)n2 * 64 + c] =
            mask[n2] ? bufA[r * 68 + c] : new_nf[(size_t)n2 * 64 + c];
    }
    wavebar();
  }
}

// ---------------------------------------------------------------------------
extern "C" void kernel_launch(void* const* d_in, const int* in_sizes, int n_in,
                              void* d_out, int out_size, void* d_ws, size_t ws_size,
                              hipStream_t stream) {
  const float* nf      = (const float*)d_in[0];
  const float* nef_out = (const float*)d_in[1];
  const float* nef_in  = (const float*)d_in[2];
  // params pytree (dict keys sorted alphabetically, each layer = (W, b)):
  // edge_mat1[2], msg_i2o[4], msg_o2i[4], net_in_trans1[2], reduce_o[4]
  int p = 3;
  const float* emW0 = (const float*)d_in[p++]; const float* emB0 = (const float*)d_in[p++];
  const float* emW1 = (const float*)d_in[p++]; const float* emB1 = (const float*)d_in[p++];
  const float* ioW0 = (const float*)d_in[p++]; const float* ioB0 = (const float*)d_in[p++];
  const float* ioW1 = (const float*)d_in[p++]; const float* ioB1 = (const float*)d_in[p++];
  const float* ioW2 = (const float*)d_in[p++]; const float* ioB2 = (const float*)d_in[p++];
  const float* ioW3 = (const float*)d_in[p++]; const float* ioB3 = (const float*)d_in[p++];
  const float* oiW0 = (const float*)d_in[p++]; const float* oiB0 = (const float*)d_in[p++];
  const float* oiW1 = (const float*)d_in[p++]; const float* oiB1 = (const float*)d_in[p++];
  const float* oiW2 = (const float*)d_in[p++]; const float* oiB2 = (const float*)d_in[p++];
  const float* oiW3 = (const float*)d_in[p++]; const float* oiB3 = (const float*)d_in[p++];
  const float* ntW0 = (const float*)d_in[p++]; const float* ntB0 = (const float*)d_in[p++];
  const float* ntW1 = (const float*)d_in[p++]; const float* ntB1 = (const float*)d_in[p++];
  const float* rW0  = (const float*)d_in[p++]; const float* rB0  = (const float*)d_in[p++];
  const float* rW1  = (const float*)d_in[p++]; const float* rB1  = (const float*)d_in[p++];
  const float* rW2  = (const float*)d_in[p++]; const float* rB2  = (const float*)d_in[p++];
  const float* rW3  = (const float*)d_in[p++]; const float* rB3  = (const float*)d_in[p++];
  const int* src_out = (const int*)d_in[p++];
  const int* dst_out = (const int*)d_in[p++];
  const int* src_in  = (const int*)d_in[p++];
  const int* dst_in  = (const int*)d_in[p++];
  const int* dri_pin = (const int*)d_in[p++];

  const int N = in_sizes[0] / 64;
  const int E = in_sizes[35];
  const int P = in_sizes[39];

  // workspace layout (floats)
  float* ws = (float*)d_ws;
  size_t off = 0;
  float* new_nf  = ws + off; off += (size_t)N * 64;
  float* att_sum = ws + off; off += (size_t)N;
  float* nfo1    = ws + off; off += (size_t)N * 4;
  float* nfo2    = ws + off; off += (size_t)N * 8;
  int*   mask    = (int*)(ws + off); off += (size_t)N;   // first 78N zeroed
  const size_t zeroN = off;
  float* node_cs = ws + off; off += (size_t)N * 4;       // fully overwritten
  float* raw_att = ws + off; off += (size_t)E;           // fully overwritten

  const int eTiles = (E + 15) / 16;
  const int nTiles = (N + 15) / 16;
  int gbE = (eTiles + 3) / 4; if (gbE > 6144) gbE = 6144;
  int gbN = (nTiles + 3) / 4; if (gbN > 6144) gbN = 6144;

  k_zero<<<4096, 256, 0, stream>>>(ws, zeroN);
  k_mask<<<(P + 255) / 256, 256, 0, stream>>>(dri_pin, mask, P);
  k_msg_o2i<<<gbE, 128, 0, stream>>>(nf, nef_out, src_out, dst_out,
                                     oiW0, oiB0, oiW1, oiB1, oiW2, oiB2, oiW3, oiB3,
                                     new_nf, E);
  k_node_trans<<<gbN, 128, 0, stream>>>(new_nf, ntW0, ntB0, ntW1, ntB1, node_cs, N);
  k_edge_att<<<gbE, 128, 0, stream>>>(new_nf, nef_in, src_in, dst_in, node_cs,
                                      emW0, emB0, emW1, emB1, raw_att, att_sum, E);
  k_msg_i2o<<<gbE, 128, 0, stream>>>(new_nf, nf, nef_in, src_in, dst_in,
                                     raw_att, att_sum,
                                     ioW0, ioB0, ioW1, ioB1, ioW2, ioB2, ioW3, ioB3,
                                     nfo1, nfo2, E);
  k_reduce<<<gbN, 128, 0, stream>>>(new_nf, nfo1, nfo2, mask,
                                    rW0, rB0, rW1, rB1, rW2, rB2, rW3, rB3,
                                    (float*)d_out, N);
}